// DCHL_26070451486837
// MI455X (gfx1250) — compile-verified
//
#include <hip/hip_runtime.h>
#include <stdint.h>

#define N_POI_   100000
#define N_EDGE_  50000
#define N_USER_  50000
#define NNZ_     1000000
#define DIMD_    128

typedef __attribute__((ext_vector_type(2))) float v2f;
typedef __attribute__((ext_vector_type(8))) float v8f;

// ---------------------------------------------------------------- zero fill
__global__ void zero_kernel(float4* __restrict__ p, int n4) {
    int i = blockIdx.x * blockDim.x + threadIdx.x;
    int stride = gridDim.x * blockDim.x;
    float4 z; z.x = z.y = z.z = z.w = 0.f;
    for (; i < n4; i += stride) p[i] = z;
}

// ------------------------------------------------- softmax of attn (4+4 vals)
__global__ void softmax_weights(const float* __restrict__ a_di,
                                const float* __restrict__ a_mv,
                                float* __restrict__ w) {
    if (threadIdx.x == 0) {
        float m = a_di[0];
        for (int i = 1; i < 4; ++i) m = fmaxf(m, a_di[i]);
        float e[4], s = 0.f;
        for (int i = 0; i < 4; ++i) { e[i] = __expf(a_di[i] - m); s += e[i]; }
        for (int i = 0; i < 4; ++i) w[i] = e[i] / s;
        m = a_mv[0];
        for (int i = 1; i < 4; ++i) m = fmaxf(m, a_mv[i]);
        s = 0.f;
        for (int i = 0; i < 4; ++i) { e[i] = __expf(a_mv[i] - m); s += e[i]; }
        for (int i = 0; i < 4; ++i) w[4 + i] = e[i] / s;
    }
}

// ------------------------------------------------------------ COO SpMM scatter
// One wave cooperatively processes 32 nonzeros: each nz's 128-float source row
// is pulled into LDS with the CDNA5 async-to-LDS path (double buffered,
// ASYNCcnt-tracked), scaled, and scattered with global_atomic_add_f32.
// NNZ is a multiple of 32 so every lane is always valid.
__global__ __launch_bounds__(256)
void spmm_scatter(const int* __restrict__ rows, const int* __restrict__ cols,
                  const float* __restrict__ vals, const float* __restrict__ x,
                  float* __restrict__ y, int nnz) {
    __shared__ float4 ldsbuf[8][2][32];          // 8 waves/block, double buffer
    const int lane = threadIdx.x & 31;
    const int wv   = threadIdx.x >> 5;
    const int wid  = (blockIdx.x * blockDim.x + threadIdx.x) >> 5;
    const int nw   = (gridDim.x * blockDim.x) >> 5;

    // low 32 bits of the generic pointer to LDS == wave-relative LDS byte addr
    const uint32_t lds0 = (uint32_t)(uintptr_t)&ldsbuf[wv][0][lane];
    const uint32_t lds1 = (uint32_t)(uintptr_t)&ldsbuf[wv][1][lane];

    for (int base = wid * 32; base < nnz; base += nw * 32) {
        const int e = base + lane;
        const int   r = rows[e];
        const int   c = cols[e];
        const float v = vals[e];

        // prefetch nz 0 into buffer 0
        {
            int cc = __shfl(c, 0, 32);
            const float4* gp = (const float4*)(x + (size_t)cc * DIMD_) + lane;
            asm volatile("global_load_async_to_lds_b128 %0, %1, off"
                         :: "v"(lds0), "v"(gp) : "memory");
        }
        #pragma unroll 1
        for (int i = 0; i < 32; ++i) {
            if (i + 1 < 32) {                    // prefetch next, wait for oldest
                int cc = __shfl(c, i + 1, 32);
                const float4* gp = (const float4*)(x + (size_t)cc * DIMD_) + lane;
                uint32_t nl = ((i + 1) & 1) ? lds1 : lds0;
                asm volatile("global_load_async_to_lds_b128 %0, %1, off"
                             :: "v"(nl), "v"(gp) : "memory");
                asm volatile("s_wait_asynccnt 0x1" ::: "memory");
            } else {
                asm volatile("s_wait_asynccnt 0x0" ::: "memory");
            }
            float4 xv = ldsbuf[wv][i & 1][lane];
            float vv  = __shfl(v, i, 32);
            int   rr  = __shfl(r, i, 32);
            float* yp = y + (size_t)rr * DIMD_ + lane * 4;
            unsafeAtomicAdd(yp + 0, xv.x * vv);
            unsafeAtomicAdd(yp + 1, xv.y * vv);
            unsafeAtomicAdd(yp + 2, xv.z * vv);
            unsafeAtomicAdd(yp + 3, xv.w * vv);
        }
    }
}

// ------------------------------------------------------- relu + residual
__global__ void relu_residual(float4* __restrict__ cur,
                              const float4* __restrict__ prev, int n4) {
    int i = blockIdx.x * blockDim.x + threadIdx.x;
    int stride = gridDim.x * blockDim.x;
    for (; i < n4; i += stride) {
        float4 m = cur[i], p = prev[i];
        m.x = fmaxf(m.x, 0.f) + p.x;
        m.y = fmaxf(m.y, 0.f) + p.y;
        m.z = fmaxf(m.z, 0.f) + p.z;
        m.w = fmaxf(m.w, 0.f) + p.w;
        cur[i] = m;
    }
}

// ----------------------------------------------- out += w * x  via WMMA f32
// 16x16 tile per wave. D = sum_b A_b(16x4) x B_b(4x16) + C, where
// A_b = x-tile columns [4b,4b+4), B_b[k,d] = (d == 4b+k) ? w : 0.
// f32 A 16x4 layout: lane L holds (M=L&15, K=2*(L>>4)+v) in VGPR v (v=0,1).
// f32 B 4x16 layout (symmetric): lane L holds (K=2*(L>>4)+v, N=L&15).
// f32 C/D 16x16: VGPR i, lane L -> (M = i + 8*(L>>4), N = L&15).
__global__ __launch_bounds__(256)
void combine_wmma(float* __restrict__ out, const float* __restrict__ x,
                  const float* __restrict__ wptr) {
    const int lane = threadIdx.x & 31;
    const int wv   = threadIdx.x >> 5;
    const int tile = blockIdx.x * 8 + wv;   // 6250 blocks * 8 waves = 50000 tiles
    const int rb = (tile >> 3) * 16;        // row tile: 0..6249
    const int cb = (tile & 7) * 16;         // col tile: 0..7
    const int m  = lane & 15;
    const int hi = lane >> 4;
    const float w = wptr[0];

    float* cptr = out + (size_t)(rb + 8 * hi) * DIMD_ + cb + m;
    v8f c;
    #pragma unroll
    for (int i = 0; i < 8; ++i) c[i] = cptr[(size_t)i * DIMD_];

    const float* xr = x + (size_t)(rb + m) * DIMD_ + cb + 2 * hi;
    #pragma unroll
    for (int b = 0; b < 4; ++b) {
        v2f A; A.x = xr[4 * b]; A.y = xr[4 * b + 1];
        v2f B;
        B.x = (m == 4 * b + 2 * hi)     ? w : 0.f;
        B.y = (m == 4 * b + 2 * hi + 1) ? w : 0.f;
        c = __builtin_amdgcn_wmma_f32_16x16x4_f32(false, A, false, B,
                                                  (short)0, c, false, false);
    }
    #pragma unroll
    for (int i = 0; i < 8; ++i) cptr[(size_t)i * DIMD_] = c[i];
}

// ---------------------------------------------------------------- launcher
extern "C" void kernel_launch(void* const* d_in, const int* in_sizes, int n_in,
                              void* d_out, int out_size, void* d_ws, size_t ws_size,
                              hipStream_t stream) {
    const float* pois     = (const float*)d_in[0];
    const int*   tar_rows = (const int*)d_in[1];
    const int*   tar_cols = (const int*)d_in[2];
    const float* tar_vals = (const float*)d_in[3];
    const int*   src_rows = (const int*)d_in[4];
    const int*   src_cols = (const int*)d_in[5];
    const float* src_vals = (const float*)d_in[6];
    const int*   up_rows  = (const int*)d_in[7];
    const int*   up_cols  = (const int*)d_in[8];
    const float* up_vals  = (const float*)d_in[9];
    const int*   pu_rows  = (const int*)d_in[10];
    const int*   pu_cols  = (const int*)d_in[11];
    const float* pu_vals  = (const float*)d_in[12];
    const float* attn_di  = (const float*)d_in[13];
    const float* attn_mv  = (const float*)d_in[14];
    float* out = (float*)d_out;

    // ws layout (floats): [0..15] softmax weights (di:0..3, mv:4..7),
    // m1: 50000*128, bufA: 100000*128, bufB: 100000*128  => ~128 MB total
    float* ws    = (float*)d_ws;
    float* w_all = ws;
    float* m1    = ws + 16;
    float* bufA  = m1 + (size_t)N_EDGE_ * DIMD_;
    float* bufB  = bufA + (size_t)N_POI_ * DIMD_;

    const int n4_poi  = N_POI_ * DIMD_ / 4;
    const int n4_edge = N_EDGE_ * DIMD_ / 4;

    softmax_weights<<<1, 64, 0, stream>>>(attn_di, attn_mv, w_all);

    // out = w_di[0]*pois + w_mv[0]*pois
    zero_kernel<<<2048, 256, 0, stream>>>((float4*)out, n4_poi);
    combine_wmma<<<6250, 256, 0, stream>>>(out, pois, w_all + 0);
    combine_wmma<<<6250, 256, 0, stream>>>(out, pois, w_all + 4);

    struct Branch {
        const int* r1; const int* c1; const float* v1; int n1;
        const int* r2; const int* c2; const float* v2; const float* w;
    };
    Branch br[2] = {
        { tar_rows, tar_cols, tar_vals, N_EDGE_, src_rows, src_cols, src_vals, w_all },
        { up_rows,  up_cols,  up_vals,  N_USER_, pu_rows,  pu_cols,  pu_vals,  w_all + 4 },
    };

    for (int bi = 0; bi < 2; ++bi) {
        const float* xprev = pois;
        float* cur = bufA; float* nxt = bufB;
        for (int l = 1; l <= 3; ++l) {
            // hop 1: edge/user aggregation
            zero_kernel<<<2048, 256, 0, stream>>>((float4*)m1, (br[bi].n1 * DIMD_) / 4);
            spmm_scatter<<<2048, 256, 0, stream>>>(br[bi].r1, br[bi].c1, br[bi].v1,
                                                   xprev, m1, NNZ_);
            // hop 2: back to POIs
            zero_kernel<<<2048, 256, 0, stream>>>((float4*)cur, n4_poi);
            spmm_scatter<<<2048, 256, 0, stream>>>(br[bi].r2, br[bi].c2, br[bi].v2,
                                                   m1, cur, NNZ_);
            // x_l = relu(m) + x_{l-1};  out += w[l] * x_l
            relu_residual<<<2048, 256, 0, stream>>>((float4*)cur, (const float4*)xprev, n4_poi);
            combine_wmma<<<6250, 256, 0, stream>>>(out, cur, br[bi].w + l);
            xprev = cur;
            float* t = cur; cur = nxt; nxt = t;
        }
    }
    (void)in_sizes; (void)n_in; (void)out_size; (void)ws_size; (void)n4_edge;
}